// DNCCell_72696616452144
// MI455X (gfx1250) — compile-verified
//
#include <hip/hip_runtime.h>
#include <hip/hip_bf16.h>
#include <math.h>

// ---------------------------------------------------------------------------
// DNC cell forward for MI455X (gfx1250, wave32).
// Memory-bound: L_prev (128MB) dominates. L_t / p_t are never materialized:
// fwd/bwd einsums fuse the L_t formula into their reductions (saves 256MB+ of
// HBM traffic; bwd's re-read of L_prev hits the 192MB L2).
// GEMMs use V_WMMA_F32_16X16X4_F32 with exact tiling (no predicated loads):
// each wave computes a 16x32 strip (2 accumulators share one A fragment).
// ---------------------------------------------------------------------------

typedef float v2f __attribute__((ext_vector_type(2)));
typedef float v8f __attribute__((ext_vector_type(8)));

constexpr int cB = 32, cN = 1024, cW = 64, cR = 4, cU = 256, cIN = 128;
constexpr int cCHI = cIN + cR * cW;            // 384
constexpr int cK0  = cCHI + cU;                // 640  (lstm0 input width)
constexpr int cK1  = cCHI + 2 * cU;            // 896  (lstm1 input width)
constexpr int cK2  = 2 * cU;                   // 512  (xi input width)
constexpr int cXI  = cR * cW + 3 * cW + 5 * cR + 3; // 471
constexpr int cXIp = 480;                      // xi padded to multiple of 32
constexpr float cEPS = 1e-8f;

__device__ __forceinline__ float sigf(float x) { return 1.0f / (1.0f + __expf(-x)); }
__device__ __forceinline__ float softplusf(float x) { return (x > 20.f) ? x : log1pf(__expf(x)); }
__device__ __forceinline__ float wredsum(float v) {
#pragma unroll
  for (int m = 16; m > 0; m >>= 1) v += __shfl_xor(v, m, 32);
  return v;
}

// ---------------------------------------------------------------------------
// f32 WMMA GEMM, exact tiling: M%16==0, N%32==0, K%4==0, no bounds checks.
// C[M,N] = A[M,K] @ B[K,N] + bias[N] (+ addm[M,N]).
// 4 waves per block; each wave owns a 16x32 output strip (two 16x16 tiles
// sharing the A fragment). A fragment is one b64 load per lane.
// ---------------------------------------------------------------------------
__global__ void k_gemm_wmma(const float* __restrict__ A, const float* __restrict__ Bm,
                            const float* __restrict__ bias, const float* __restrict__ addm,
                            float* __restrict__ C,
                            int M, int N, int K, int lda, int ldb, int ldc) {
  const int ngrp  = N >> 5;                 // 32-wide column groups
  const int npair = (M >> 4) * ngrp;
  const int wv    = threadIdx.x >> 5;
  const int pair  = blockIdx.x * 4 + wv;
  if (pair >= npair) return;                // wave-uniform: EXEC stays all-ones
  const int mt   = pair / ngrp;
  const int ng   = pair % ngrp;
  const int lane = threadIdx.x & 31;
  const int half = lane >> 4;               // K-half select for A/B fragments
  const int l16  = lane & 15;
  const int row0 = mt << 4;
  const int col0 = ng << 5;
  const int arow = row0 + l16;
  const int bc0  = col0 + l16;
  const int bc1  = bc0 + 16;

  v8f acc0, acc1;
  const float bi0 = bias ? bias[bc0] : 0.0f;
  const float bi1 = bias ? bias[bc1] : 0.0f;
#pragma unroll
  for (int j = 0; j < 8; ++j) { acc0[j] = bi0; acc1[j] = bi1; }

  const float* Arow = A + (size_t)arow * lda;
#pragma unroll 4
  for (int k = 0; k < K; k += 4) {
    const int ka = k + (half << 1);         // lanes 0-15: K={k,k+1}; 16-31: {k+2,k+3}
    v2f a = *(const v2f*)(Arow + ka);       // single b64 load (8B-aligned: ka even)
    const float* Bk = Bm + (size_t)ka * ldb;
    v2f b0, b1;
    b0.x = Bk[bc0];        b1.x = Bk[bc1];
    b0.y = Bk[ldb + bc0];  b1.y = Bk[ldb + bc1];
    acc0 = __builtin_amdgcn_wmma_f32_16x16x4_f32(false, a, false, b0, (short)0, acc0,
                                                 false, false);
    acc1 = __builtin_amdgcn_wmma_f32_16x16x4_f32(false, a, false, b1, (short)0, acc1,
                                                 false, false);
  }

#pragma unroll
  for (int j = 0; j < 8; ++j) {
    const int r = row0 + (half << 3) + j;   // C layout: vgpr j -> M=j (+8 for hi lanes)
    float v0 = acc0[j], v1 = acc1[j];
    if (addm) { v0 += addm[r * ldc + bc0]; v1 += addm[r * ldc + bc1]; }
    C[r * ldc + bc0] = v0;
    C[r * ldc + bc1] = v1;
  }
}

// Zero-pad Wxi (512x471 -> 512x480) and bxi (471 -> 480) so the xi GEMM tiles
// exactly (removes all predication from the WMMA hot loop).
__global__ void k_padw(const float* __restrict__ Wxi, const float* __restrict__ bxi,
                       float* __restrict__ Wp, float* __restrict__ bp) {
  int t = blockIdx.x * blockDim.x + threadIdx.x;
  if (t < cXIp) bp[t] = (t < cXI) ? bxi[t] : 0.0f;
  if (t >= cK2 * cXIp) return;
  int k = t / cXIp, c = t % cXIp;
  Wp[t] = (c < cXI) ? Wxi[k * cXI + c] : 0.0f;
}

// --------------------------- small fused kernels ---------------------------

// Build in0 = [x_t | R_prev^T | H0] (B x 640) and in1[:, :640] = [chi | H1].
__global__ void k_prep(const float* __restrict__ x_t, const float* __restrict__ Rp,
                       const float* __restrict__ Hp, float* __restrict__ in0,
                       float* __restrict__ in1) {
  int t = blockIdx.x * blockDim.x + threadIdx.x;
  if (t >= cB * cK0) return;
  int b = t / cK0, c = t % cK0;
  if (c < cCHI) {
    float val;
    if (c < cIN) val = x_t[b * cIN + c];
    else {
      int q = c - cIN;
      val = Rp[(((q >> 6) * cB) + b) * cW + (q & 63)];  // R_prev (R,B,W) -> (B, R*W)
    }
    in0[b * cK0 + c] = val;
    in1[b * cK1 + c] = val;
  } else {
    int u = c - cCHI;
    in0[b * cK0 + c] = Hp[(0 * cB + b) * cU + u];
    in1[b * cK1 + c] = Hp[(1 * cB + b) * cU + u];
  }
}

// LSTM gates: h = sig(o)*tanh(sig(f)*s_prev + sig(i)*tanh(u)); s_prev==null -> 0.
__global__ void k_gates(const float* __restrict__ g, const float* __restrict__ s_prev,
                        float* __restrict__ dstA, int strideA,
                        float* __restrict__ dstB, int strideB) {
  int t = blockIdx.x * blockDim.x + threadIdx.x;
  if (t >= cB * cU) return;
  int b = t / cU, u = t % cU;
  const float* gr = g + b * 4 * cU;
  float i = sigf(gr[u]);
  float f = sigf(gr[cU + u]);
  float uu = tanhf(gr[2 * cU + u]);
  float o = sigf(gr[3 * cU + u]);
  float s = i * uu;
  if (s_prev) s += f * s_prev[b * cU + u];
  float h = o * tanhf(s);
  dstA[b * strideA + u] = h;
  if (dstB) dstB[b * strideB + u] = h;
}

// Parse xi (B x 480-padded, 471 valid) into head parameters.
__global__ void k_heads(const float* __restrict__ xi, float* Kr, float* Betar,
                        float* kw, float* betaw, float* e, float* v, float* F,
                        float* ga, float* gw, float* PI) {
  int t = blockIdx.x * blockDim.x + threadIdx.x;
  if (t >= cB * cXI) return;
  int b = t / cXI, c = t % cXI;
  float x = xi[b * cXIp + c];
  if (c < 256)        { Kr[(((c >> 6) * cB) + b) * cW + (c & 63)] = x; }
  else if (c < 260)   { Betar[(c - 256) * cB + b] = 1.0f + softplusf(x); }
  else if (c < 324)   { kw[b * cW + (c - 260)] = x; }
  else if (c == 324)  { betaw[b] = 1.0f + softplusf(x); }
  else if (c < 389)   { e[b * cW + (c - 325)] = sigf(x); }
  else if (c < 453)   { v[b * cW + (c - 389)] = x; }
  else if (c < 457)   { F[b * cR + (c - 453)] = sigf(x); }
  else if (c == 457)  { ga[b] = sigf(x); }
  else if (c == 458)  { gw[b] = sigf(x); }
  else {                                              // PI: softmax over groups of 3
    int q = c - 459;
    if (q % 3 == 0) {
      int r = q / 3;
      float a0 = x, a1 = xi[b * cXIp + c + 1], a2 = xi[b * cXIp + c + 2];
      float mx = fmaxf(a0, fmaxf(a1, a2));
      float e0 = __expf(a0 - mx), e1 = __expf(a1 - mx), e2 = __expf(a2 - mx);
      float s = e0 + e1 + e2;
      float* p = PI + ((r * cB) + b) * 3;              // (R,B,3)
      p[0] = e0 / s; p[1] = e1 / s; p[2] = e2 / s;
    }
  }
}

// psi = prod_r(1 - Wr_prev*F);  u_t = (u + ww - u*ww)*psi.
__global__ void k_usage(const float* __restrict__ Wrp, const float* __restrict__ F,
                        const float* __restrict__ up, const float* __restrict__ wwp,
                        float* __restrict__ ut) {
  int t = blockIdx.x * blockDim.x + threadIdx.x;
  if (t >= cB * cN) return;
  int b = t / cN, n = t % cN;
  float psi = 1.0f;
#pragma unroll
  for (int r = 0; r < cR; ++r) psi *= 1.0f - Wrp[((r * cB) + b) * cN + n] * F[b * cR + r];
  float u = up[t], w = wwp[t];
  ut[t] = (u + w - u * w) * psi;
}

// Allocation: bitonic argsort (1024 keys) + log-time product scan, all in LDS.
__global__ void k_alloc(const float* __restrict__ ut, float* __restrict__ at) {
  __shared__ float key[cN];
  __shared__ int   idx[cN];
  const int b = blockIdx.x, tid = threadIdx.x;
  key[tid] = ut[b * cN + tid];
  idx[tid] = tid;
  __syncthreads();
  for (int k = 2; k <= cN; k <<= 1) {
    for (int j = k >> 1; j > 0; j >>= 1) {
      int ixj = tid ^ j;
      if (ixj > tid) {
        float k1 = key[tid], k2 = key[ixj];
        int   i1 = idx[tid], i2 = idx[ixj];
        bool up = ((tid & k) == 0);
        bool gt = (k1 > k2) || (k1 == k2 && i1 > i2);   // stable tiebreak
        if (up == gt) { key[tid] = k2; key[ixj] = k1; idx[tid] = i2; idx[ixj] = i1; }
      }
      __syncthreads();
    }
  }
  float su  = key[tid];       // sorted usage
  int   dst = idx[tid];       // original position
  // inclusive product scan (Hillis-Steele)
  for (int off = 1; off < cN; off <<= 1) {
    float tpre = (tid >= off) ? key[tid - off] : 1.0f;
    __syncthreads();
    key[tid] *= tpre;
    __syncthreads();
  }
  float ex = (tid > 0) ? key[tid - 1] : 1.0f;           // exclusive cumprod
  at[b * cN + dst] = (1.0f - su) * ex;
}

// Content weighting c_w = softmax(beta * cosine(M_prev, k_w)).
__global__ void k_cw(const float* __restrict__ Mp, const float* __restrict__ kw,
                     const float* __restrict__ betaw, float* __restrict__ cw) {
  __shared__ float sims[cN];
  __shared__ float red[256];
  const int b = blockIdx.x, tid = threadIdx.x;
  const int lane = tid & 31, wv = tid >> 5;
  const float k1 = kw[b * cW + lane], k2 = kw[b * cW + 32 + lane];
  const float nk = sqrtf(wredsum(k1 * k1 + k2 * k2));
  for (int n = wv; n < cN; n += 8) {                    // wave per row, lanes over W
    const float* mrow = Mp + ((size_t)b * cN + n) * cW;
    float m1 = mrow[lane], m2 = mrow[32 + lane];
    float d  = wredsum(m1 * k1 + m2 * k2);
    float nm = sqrtf(wredsum(m1 * m1 + m2 * m2));
    if (lane == 0) sims[n] = d / fmaxf(nm * nk, cEPS);
  }
  __syncthreads();
  const float beta = betaw[b];
  float lmax = -3.4e38f;
  for (int n = tid; n < cN; n += 256) lmax = fmaxf(lmax, sims[n] * beta);
  red[tid] = lmax; __syncthreads();
  for (int s = 128; s > 0; s >>= 1) { if (tid < s) red[tid] = fmaxf(red[tid], red[tid + s]); __syncthreads(); }
  const float mx = red[0]; __syncthreads();
  float lsum = 0.f;
  for (int n = tid; n < cN; n += 256) lsum += __expf(sims[n] * beta - mx);
  red[tid] = lsum; __syncthreads();
  for (int s = 128; s > 0; s >>= 1) { if (tid < s) red[tid] += red[tid + s]; __syncthreads(); }
  const float inv = 1.0f / red[0];
  for (int n = tid; n < cN; n += 256) cw[b * cN + n] = __expf(sims[n] * beta - mx) * inv;
}

// w_w = g_w * (g_a * a_t + (1-g_a) * c_w)
__global__ void k_ww(const float* __restrict__ at, const float* __restrict__ cw,
                     const float* __restrict__ ga, const float* __restrict__ gw,
                     float* __restrict__ ww) {
  int t = blockIdx.x * blockDim.x + threadIdx.x;
  if (t >= cB * cN) return;
  int b = t / cN;
  float a = ga[b];
  ww[t] = gw[b] * (a * at[t] + (1.0f - a) * cw[t]);
}

// M_t = M*(1 - w_w*e) + w_w*v ; nM = ||M_t|| per row. Block = one (b,n), 64 thr.
__global__ void k_mem(const float* __restrict__ Mp, const float* __restrict__ ww,
                      const float* __restrict__ e, const float* __restrict__ v,
                      float* __restrict__ Mt, float* __restrict__ nM) {
  __shared__ float part[2];
  const int bn = blockIdx.x;            // b*N + n
  const int b = bn >> 10;
  const int w = threadIdx.x;
  const float wwv = ww[bn];
  const float m  = Mp[(size_t)bn * cW + w];
  const float mt = m * (1.0f - wwv * e[b * cW + w]) + wwv * v[b * cW + w];
  Mt[(size_t)bn * cW + w] = mt;
  float s = wredsum(mt * mt);
  if ((w & 31) == 0) part[w >> 5] = s;
  __syncthreads();
  if (w == 0) nM[bn] = sqrtf(part[0] + part[1]);
}

// fwd[r,b,i] = sum_j L_t(i,j)*Wr[r,b,j]; L_t computed on the fly (never stored).
__global__ void k_fwd(const float* __restrict__ Lp, const float* __restrict__ ww,
                      const float* __restrict__ pp, const float* __restrict__ Wrp,
                      float* __restrict__ fwd) {
  __shared__ float lww[cN], lpp[cN], lwr[cR][cN];      // 24KB
  const int b = blockIdx.x, tid = threadIdx.x;
  for (int t = tid; t < cN; t += 256) {
    lww[t] = ww[b * cN + t];
    lpp[t] = pp[b * cN + t];
#pragma unroll
    for (int r = 0; r < cR; ++r) lwr[r][t] = Wrp[((r * cB) + b) * cN + t];
  }
  __syncthreads();
  const int lane = tid & 31, wv = tid >> 5;
  const int i = blockIdx.y * 8 + wv;                   // one wave per L row
  const float wwi = lww[i];
  const float* Lrow = Lp + ((size_t)b * cN + i) * cN;
  float a0 = 0, a1 = 0, a2 = 0, a3 = 0;
  for (int j0 = 0; j0 < cN; j0 += 32) {
    if (j0 + 256 < cN) __builtin_prefetch(Lrow + j0 + 256, 0, 0);
    const int jj = j0 + lane;
    float lp = Lrow[jj];
    float lt = (jj == i) ? 0.0f
                         : ((1.0f - wwi - lww[jj]) * lp + wwi * lpp[jj]);
    a0 += lt * lwr[0][jj]; a1 += lt * lwr[1][jj];
    a2 += lt * lwr[2][jj]; a3 += lt * lwr[3][jj];
  }
  a0 = wredsum(a0); a1 = wredsum(a1); a2 = wredsum(a2); a3 = wredsum(a3);
  if (lane == 0) {
    fwd[(0 * cB + b) * cN + i] = a0; fwd[(1 * cB + b) * cN + i] = a1;
    fwd[(2 * cB + b) * cN + i] = a2; fwd[(3 * cB + b) * cN + i] = a3;
  }
}

// bwd[r,b,i] = sum_j L_t(j,i)*Wr[r,b,j]; thread-per-column, coalesced rows.
__global__ void k_bwd(const float* __restrict__ Lp, const float* __restrict__ ww,
                      const float* __restrict__ pp, const float* __restrict__ Wrp,
                      float* __restrict__ bwd) {
  __shared__ float lww[cN], lwr[cR][cN];               // 20KB
  const int b = blockIdx.x, tid = threadIdx.x;
  const int i = blockIdx.y * 256 + tid;
  for (int t = tid; t < cN; t += 256) {
    lww[t] = ww[b * cN + t];
#pragma unroll
    for (int r = 0; r < cR; ++r) lwr[r][t] = Wrp[((r * cB) + b) * cN + t];
  }
  __syncthreads();
  const float wwi = lww[i];
  const float ppi = pp[b * cN + i];
  const float* Lb = Lp + (size_t)b * cN * cN;
  float a0 = 0, a1 = 0, a2 = 0, a3 = 0;
  for (int j = 0; j < cN; ++j) {
    float lp = Lb[(size_t)j * cN + i];
    float wwj = lww[j];
    float lt = (j == i) ? 0.0f : ((1.0f - wwj - wwi) * lp + wwj * ppi);
    a0 += lt * lwr[0][j]; a1 += lt * lwr[1][j];
    a2 += lt * lwr[2][j]; a3 += lt * lwr[3][j];
  }
  bwd[(0 * cB + b) * cN + i] = a0; bwd[(1 * cB + b) * cN + i] = a1;
  bwd[(2 * cB + b) * cN + i] = a2; bwd[(3 * cB + b) * cN + i] = a3;
}

// c_r softmax fused with W_r = PI0*bwd + PI1*c_r + PI2*fwd.  Block per (r,b).
__global__ void k_readw(const float* __restrict__ Mt, const float* __restrict__ nM,
                        const float* __restrict__ Kr, const float* __restrict__ Betar,
                        const float* __restrict__ PI, const float* __restrict__ fwd,
                        const float* __restrict__ bwd, float* __restrict__ Wr) {
  __shared__ float sims[cN];
  __shared__ float red[256];
  const int rb = blockIdx.x;           // r*32 + b
  const int b  = rb & 31;
  const int tid = threadIdx.x, lane = tid & 31, wv = tid >> 5;
  const float k1 = Kr[rb * cW + lane], k2 = Kr[rb * cW + 32 + lane];
  const float nk = sqrtf(wredsum(k1 * k1 + k2 * k2));
  const float beta = Betar[rb];
  for (int n = wv; n < cN; n += 8) {
    const float* mrow = Mt + ((size_t)b * cN + n) * cW;
    float m1 = mrow[lane], m2 = mrow[32 + lane];
    float d = wredsum(m1 * k1 + m2 * k2);
    if (lane == 0) sims[n] = beta * (d / fmaxf(nM[b * cN + n] * nk, cEPS));
  }
  __syncthreads();
  float lmax = -3.4e38f;
  for (int n = tid; n < cN; n += 256) lmax = fmaxf(lmax, sims[n]);
  red[tid] = lmax; __syncthreads();
  for (int s = 128; s > 0; s >>= 1) { if (tid < s) red[tid] = fmaxf(red[tid], red[tid + s]); __syncthreads(); }
  const float mx = red[0]; __syncthreads();
  float lsum = 0.f;
  for (int n = tid; n < cN; n += 256) lsum += __expf(sims[n] - mx);
  red[tid] = lsum; __syncthreads();
  for (int s = 128; s > 0; s >>= 1) { if (tid < s) red[tid] += red[tid + s]; __syncthreads(); }
  const float inv = 1.0f / red[0];
  const float pi0 = PI[rb * 3 + 0], pi1 = PI[rb * 3 + 1], pi2 = PI[rb * 3 + 2];
  for (int n = tid; n < cN; n += 256) {
    float cr = __expf(sims[n] - mx) * inv;
    Wr[rb * cN + n] = pi0 * bwd[rb * cN + n] + pi1 * cr + pi2 * fwd[rb * cN + n];
  }
}

// R_t: Rflat[b, r*W+w] = sum_n M_t[b,n,w] * W_r[r,b,n].  Block per (r,b), 64 thr.
__global__ void k_readvec(const float* __restrict__ Mt, const float* __restrict__ Wr,
                          float* __restrict__ Rf) {
  const int rb = blockIdx.x, r = rb >> 5, b = rb & 31;
  const int w = threadIdx.x;
  float acc = 0.f;
  for (int n = 0; n < cN; ++n)
    acc += Mt[((size_t)b * cN + n) * cW + w] * Wr[rb * cN + n];
  Rf[b * (cR * cW) + r * cW + w] = acc;
}

// ---------------------------------------------------------------------------

extern "C" void kernel_launch(void* const* d_in, const int* in_sizes, int n_in,
                              void* d_out, int out_size, void* d_ws, size_t ws_size,
                              hipStream_t stream) {
  const float* x_t   = (const float*)d_in[0];
  const float* Rp    = (const float*)d_in[1];
  const float* Hp    = (const float*)d_in[2];
  const float* Sp    = (const float*)d_in[3];
  const float* Mp    = (const float*)d_in[4];
  const float* up    = (const float*)d_in[5];
  const float* Wrp   = (const float*)d_in[6];
  const float* wwp   = (const float*)d_in[7];
  const float* pp    = (const float*)d_in[8];
  const float* Lp    = (const float*)d_in[9];
  const float* Wl0   = (const float*)d_in[10];
  const float* bl0   = (const float*)d_in[11];
  const float* Wl1   = (const float*)d_in[12];
  const float* bl1   = (const float*)d_in[13];
  const float* Wxi   = (const float*)d_in[14];
  const float* bxi   = (const float*)d_in[15];
  const float* Wread = (const float*)d_in[16];
  const float* bread = (const float*)d_in[17];

  float* ws = (float*)d_ws;
  size_t o = 0;
  float* in0   = ws + o; o += (size_t)cB * cK0;
  float* in1   = ws + o; o += (size_t)cB * cK1;
  float* in2   = ws + o; o += (size_t)cB * cK2;
  float* g0    = ws + o; o += (size_t)cB * 4 * cU;
  float* g1    = ws + o; o += (size_t)cB * 4 * cU;
  float* xi    = ws + o; o += (size_t)cB * cXIp;       // padded ldc = 480
  float* Wxip  = ws + o; o += (size_t)cK2 * cXIp;      // zero-padded Wxi
  float* bxip  = ws + o; o += (size_t)cXIp;
  float* Kr    = ws + o; o += (size_t)cR * cB * cW;
  float* Betar = ws + o; o += (size_t)cR * cB;
  float* kw    = ws + o; o += (size_t)cB * cW;
  float* betaw = ws + o; o += (size_t)cB;
  float* eb    = ws + o; o += (size_t)cB * cW;
  float* vb    = ws + o; o += (size_t)cB * cW;
  float* Fb    = ws + o; o += (size_t)cB * cR;
  float* ga    = ws + o; o += (size_t)cB;
  float* gw    = ws + o; o += (size_t)cB;
  float* PI    = ws + o; o += (size_t)cR * cB * 3 + 1; // +1 makes the later offsets odd
  float* ut    = ws + o; o += (size_t)cB * cN;
  float* at    = ws + o; o += (size_t)cB * cN;
  float* cw    = ws + o; o += (size_t)cB * cN;
  float* wwb   = ws + o; o += (size_t)cB * cN;
  float* Mt    = ws + o; o += (size_t)cB * cN * cW;    // 8 MB
  float* nM    = ws + o; o += (size_t)cB * cN;
  float* fwdb  = ws + o; o += (size_t)cR * cB * cN;
  float* bwdb  = ws + o; o += (size_t)cR * cB * cN;
  float* Wrb   = ws + o; o += (size_t)cR * cB * cN;
  float* Rf    = ws + o; o += (size_t)cB * cR * cW;

  auto gemmBlocks = [](int M, int N) {                 // 16x32 strip per wave, 4 waves/blk
    int npair = (M >> 4) * (N >> 5);
    return (npair + 3) / 4;
  };

  // 0. Pad Wxi/bxi for exact tiling
  k_padw<<<(cK2 * cXIp + 255) / 256, 256, 0, stream>>>(Wxi, bxi, Wxip, bxip);
  // 1. Build LSTM inputs
  k_prep<<<(cB * cK0 + 255) / 256, 256, 0, stream>>>(x_t, Rp, Hp, in0, in1);
  // 2. LSTM0 GEMM (32x640 @ 640x1024) + gates
  k_gemm_wmma<<<gemmBlocks(cB, 4 * cU), 128, 0, stream>>>(in0, Wl0, bl0, nullptr, g0,
                                                          cB, 4 * cU, cK0, cK0, 4 * cU, 4 * cU);
  k_gates<<<(cB * cU + 255) / 256, 256, 0, stream>>>(g0, Sp, in1 + cK0, cK1, in2, cK2);
  // 3. LSTM1 GEMM (32x896 @ 896x1024) + gates (s_prev = 0)
  k_gemm_wmma<<<gemmBlocks(cB, 4 * cU), 128, 0, stream>>>(in1, Wl1, bl1, nullptr, g1,
                                                          cB, 4 * cU, cK1, cK1, 4 * cU, 4 * cU);
  k_gates<<<(cB * cU + 255) / 256, 256, 0, stream>>>(g1, nullptr, in2 + cU, cK2, nullptr, 0);
  // 4. xi GEMM (32x512 @ 512x480-padded)
  k_gemm_wmma<<<gemmBlocks(cB, cXIp), 128, 0, stream>>>(in2, Wxip, bxip, nullptr, xi,
                                                        cB, cXIp, cK2, cK2, cXIp, cXIp);
  // 5. Parse heads
  k_heads<<<(cB * cXI + 255) / 256, 256, 0, stream>>>(xi, Kr, Betar, kw, betaw,
                                                      eb, vb, Fb, ga, gw, PI);
  // 6. Usage, allocation (bitonic sort), content weighting, write weights
  k_usage<<<(cB * cN + 255) / 256, 256, 0, stream>>>(Wrp, Fb, up, wwp, ut);
  k_alloc<<<cB, cN, 0, stream>>>(ut, at);
  k_cw<<<cB, 256, 0, stream>>>(Mp, kw, betaw, cw);
  k_ww<<<(cB * cN + 255) / 256, 256, 0, stream>>>(at, cw, ga, gw, wwb);
  // 7. Memory update + row norms
  k_mem<<<cB * cN, cW, 0, stream>>>(Mp, wwb, eb, vb, Mt, nM);
  // 8. fwd/bwd with L_t fused (L_prev streamed once; bwd re-read is L2-resident)
  { dim3 gf(cB, cN / 8);   k_fwd<<<gf, 256, 0, stream>>>(Lp, wwb, pp, Wrp, fwdb); }
  { dim3 gb(cB, cN / 256); k_bwd<<<gb, 256, 0, stream>>>(Lp, wwb, pp, Wrp, bwdb); }
  // 9. Read weights (c_r softmax fused with PI mix)
  k_readw<<<cR * cB, 256, 0, stream>>>(Mt, nM, Kr, Betar, PI, fwdb, bwdb, Wrb);
  // 10. Read vectors
  k_readvec<<<cR * cB, cW, 0, stream>>>(Mt, Wrb, Rf);
  // 11. y = v + Rflat @ Wread + bread  (WMMA, fused add)
  k_gemm_wmma<<<gemmBlocks(cB, cW), 128, 0, stream>>>(Rf, Wread, bread, vb, (float*)d_out,
                                                      cB, cW, cR * cW, cR * cW, cW, cW);
}